// Model_987842478282
// MI455X (gfx1250) — compile-verified
//
#include <hip/hip_runtime.h>

typedef __attribute__((ext_vector_type(2))) float v2f;
typedef __attribute__((ext_vector_type(8))) float v8f;

#define BATCH 8
#define CIN   3
#define HH    512
#define WW    512
#define HO    510
#define WO    510
#define KTAP  9             // 3x3 taps
#define K27   27            // CIN*KTAP
#define NTOT  (BATCH*HO*WO) // 2,080,800 output pixels

// Fused: offset-conv (im2col GEMM via f32 WMMA) -> LDS -> deformable conv gather.
// Workgroup = 256 threads = 8 waves; each wave owns 32 pixels (2 N-tiles of 16).
__global__ __launch_bounds__(256)
void deform_fused_kernel(const float* __restrict__ x,
                         const float* __restrict__ w1,
                         const float* __restrict__ b1,
                         const float* __restrict__ w2,
                         const float* __restrict__ b2,
                         float* __restrict__ out)
{
    __shared__ float lds_off[8][32][18]; // per-wave offsets: 18 KB (stride-18 is conflict-free)
    __shared__ float lds_w2[81];
    __shared__ float lds_b1[18];
    __shared__ float lds_b2[3];

    const int tid  = threadIdx.x;
    const int wave = tid >> 5;
    const int lane = tid & 31;
    const int nl   = lane & 15;            // N-column within 16-wide tile
    const int kh   = (lane & 16) ? 2 : 0;  // K-half select for A/B fragments

    if (tid < 81) lds_w2[tid] = w2[tid];
    if (tid < 18) lds_b1[tid] = b1[tid];
    if (tid < 3)  lds_b2[tid] = b2[tid];

    // ---- A fragments: W1 as [18][27] row-major, padded to 32x28 ----
    // A 16x4 layout: lane<16 holds {A[m][k0],A[m][k0+1]}, lane>=16 holds {A[m][k0+2],A[m][k0+3]}
    // ---- plus per-lane im2col address deltas (hoisted out of the pixel loop) ----
    v2f a0[7], a1[7];
    int d0[7], d1[7];
    #pragma unroll
    for (int t = 0; t < 7; ++t) {
        int k  = t * 4 + kh;
        int m0 = nl;        // rows 0..15
        int m1 = 16 + nl;   // rows 16..31 (only 16,17 live)
        float a00 = (k     < K27) ? w1[m0 * K27 + k]     : 0.0f;
        float a01 = (k + 1 < K27) ? w1[m0 * K27 + k + 1] : 0.0f;
        v2f ta; ta[0] = a00; ta[1] = a01; a0[t] = ta;
        float a10 = (m1 < 18 && k     < K27) ? w1[m1 * K27 + k]     : 0.0f;
        float a11 = (m1 < 18 && k + 1 < K27) ? w1[m1 * K27 + k + 1] : 0.0f;
        v2f tb; tb[0] = a10; tb[1] = a11; a1[t] = tb;

        // Clamp k to 26 for the single K=27 pad: A is zero there, so B may be
        // any in-bounds value — no masking, no divergence.
        int kc0 = (k     < K27) ? k     : K27 - 1;
        int kc1 = (k + 1 < K27) ? k + 1 : K27 - 1;
        d0[t] = (kc0 / KTAP) * (HH * WW) + ((kc0 % KTAP) / 3) * WW + (kc0 % 3);
        d1[t] = (kc1 / KTAP) * (HH * WW) + ((kc1 % KTAP) / 3) * WW + (kc1 % 3);
    }

    __syncthreads(); // lds_b1 ready before offset stores

    const int wg_base = blockIdx.x * 256;

    // ---- Stage 1: offset GEMM, 2 tiles of 16 pixels per wave, branch-free ----
    #pragma unroll
    for (int tile = 0; tile < 2; ++tile) {
        int n  = wg_base + wave * 32 + tile * 16 + nl;  // pixel for this column
        int nc = (n < NTOT) ? n : (NTOT - 1);           // clamp: tail lanes load valid
                                                        // addresses, results unread
        int b  = nc / (HO * WO);
        int r  = nc % (HO * WO);
        int oy = r / WO;
        int ox = r % WO;
        int base = (b * CIN * HH + oy) * WW + ox;       // x[b, 0, oy, ox]

        v8f c0 = {}; v8f c1 = {};
        #pragma unroll
        for (int t = 0; t < 7; ++t) {
            v2f bf;
            bf[0] = x[base + d0[t]];
            bf[1] = x[base + d1[t]];
            c0 = __builtin_amdgcn_wmma_f32_16x16x4_f32(false, a0[t], false, bf,
                                                       (short)0, c0, false, false);
            c1 = __builtin_amdgcn_wmma_f32_16x16x4_f32(false, a1[t], false, bf,
                                                       (short)0, c1, false, false);
        }

        // D layout: lane owns column nl, rows v + (lane<16 ? 0 : 8)
        int pxl = tile * 16 + nl;
        #pragma unroll
        for (int v = 0; v < 8; ++v) {
            int m = v + ((lane & 16) ? 8 : 0);   // channel 0..15
            lds_off[wave][pxl][m] = c0[v] + lds_b1[m];
        }
        if (!(lane & 16)) {                      // channels 16,17 from M-tile-1 rows 0,1
            lds_off[wave][pxl][16] = c1[0] + lds_b1[16];
            lds_off[wave][pxl][17] = c1[1] + lds_b1[17];
        }
    }

    __syncthreads();

    // ---- Stage 2: deformable gather, one pixel per lane ----
    int n = wg_base + wave * 32 + lane;
    if (n < NTOT) {
        int b  = n / (HO * WO);
        int r  = n % (HO * WO);
        int oy = r / WO;
        int ox = r % WO;
        const float* off = &lds_off[wave][lane][0];

        float acc0 = lds_b2[0], acc1 = lds_b2[1], acc2 = lds_b2[2];
        #pragma unroll
        for (int k = 0; k < KTAP; ++k) {
            float dy = off[2 * k];
            float dx = off[2 * k + 1];
            float py = (float)(oy + k / 3) + dy;
            float px = (float)(ox + k % 3) + dx;
            float valid = (py > -1.0f && py < (float)HH &&
                           px > -1.0f && px < (float)WW) ? 1.0f : 0.0f;
            float y0f = floorf(py), x0f = floorf(px);
            float ly = py - y0f, lx = px - x0f;
            int y0 = (int)y0f, x0i = (int)x0f;
            float w00 = (1.0f - ly) * (1.0f - lx);
            float w01 = (1.0f - ly) * lx;
            float w10 = ly * (1.0f - lx);
            float w11 = ly * lx;
            bool iy0 = (y0 >= 0 && y0 < HH);
            bool iy1 = (y0 + 1 >= 0 && y0 + 1 < HH);
            bool ix0 = (x0i >= 0 && x0i < WW);
            bool ix1 = (x0i + 1 >= 0 && x0i + 1 < WW);
            int cy0 = min(max(y0, 0), HH - 1);
            int cy1 = min(max(y0 + 1, 0), HH - 1);
            int cx0 = min(max(x0i, 0), WW - 1);
            int cx1 = min(max(x0i + 1, 0), WW - 1);
            #pragma unroll
            for (int c = 0; c < CIN; ++c) {
                const float* xb = x + (size_t)(b * CIN + c) * HH * WW;
                float v00 = (iy0 && ix0) ? xb[cy0 * WW + cx0] : 0.0f;
                float v01 = (iy0 && ix1) ? xb[cy0 * WW + cx1] : 0.0f;
                float v10 = (iy1 && ix0) ? xb[cy1 * WW + cx0] : 0.0f;
                float v11 = (iy1 && ix1) ? xb[cy1 * WW + cx1] : 0.0f;
                float s = (v00 * w00 + v01 * w01 + v10 * w10 + v11 * w11) * valid;
                acc0 += s * lds_w2[(0 * CIN + c) * KTAP + k];
                acc1 += s * lds_w2[(1 * CIN + c) * KTAP + k];
                acc2 += s * lds_w2[(2 * CIN + c) * KTAP + k];
            }
        }
        size_t obase = ((size_t)b * 3) * HO * WO + (size_t)oy * WO + ox;
        out[obase]                       = acc0;
        out[obase + (size_t)HO * WO]     = acc1;
        out[obase + (size_t)2 * HO * WO] = acc2;
    }
}

extern "C" void kernel_launch(void* const* d_in, const int* in_sizes, int n_in,
                              void* d_out, int out_size, void* d_ws, size_t ws_size,
                              hipStream_t stream) {
    (void)in_sizes; (void)n_in; (void)d_ws; (void)ws_size; (void)out_size;
    const float* x  = (const float*)d_in[0];
    const float* w1 = (const float*)d_in[1];
    const float* b1 = (const float*)d_in[2];
    const float* w2 = (const float*)d_in[3];
    const float* b2 = (const float*)d_in[4];
    float* out = (float*)d_out;

    int nblocks = (NTOT + 255) / 256;
    hipLaunchKernelGGL(deform_fused_kernel, dim3(nblocks), dim3(256), 0, stream,
                       x, w1, b1, w2, b2, out);
}